// GCNIIwithJK_58480274703251
// MI455X (gfx1250) — compile-verified
//
#include <hip/hip_runtime.h>
#include <math.h>

typedef float v2f __attribute__((ext_vector_type(2)));
typedef float v8f __attribute__((ext_vector_type(8)));

#define HCH 128              // hidden channels
#define RPB 64               // rows per block (4 waves x 16)
#define TPB 128              // threads per block
#define LDS_BYTES (((size_t)HCH * HCH + (size_t)RPB * HCH) * sizeof(float))

// ---------------- utility kernels ----------------

__global__ void k_fill(float* __restrict__ p, float v, long n) {
  long i = (long)blockIdx.x * blockDim.x + threadIdx.x;
  long stride = (long)gridDim.x * blockDim.x;
  for (; i < n; i += stride) p[i] = v;
}

__global__ void k_deg_count(const int* __restrict__ dst, float* __restrict__ deg, int E) {
  int e = blockIdx.x * blockDim.x + threadIdx.x;
  if (e < E) atomicAdd(&deg[dst[e]], 1.0f);
}

__global__ void k_rsqrt(float* __restrict__ p, int n) {
  int i = blockIdx.x * blockDim.x + threadIdx.x;
  if (i < n) p[i] = rsqrtf(p[i]);
}

// z[dst] += h[src] * dinv[src]*dinv[dst]; one wave per edge, each lane does 4 channels
__global__ void k_edge_gcn(const int* __restrict__ src, const int* __restrict__ dst,
                           const float* __restrict__ dinv, const float* __restrict__ h,
                           float* __restrict__ z, int E) {
  int wid = (int)(((long)blockIdx.x * blockDim.x + threadIdx.x) >> 5);
  int lane = threadIdx.x & 31;
  if (wid >= E) return;
  int s = src[wid], d = dst[wid];
  float coef = dinv[s] * dinv[d];
  float4 hv = ((const float4*)(h + (long)s * HCH))[lane];
  float* zd = z + (long)d * HCH + lane * 4;
  atomicAdd(zd + 0, hv.x * coef);
  atomicAdd(zd + 1, hv.y * coef);
  atomicAdd(zd + 2, hv.z * coef);
  atomicAdd(zd + 3, hv.w * coef);
}

// agg[dst] += z[src]; one wave per edge
__global__ void k_edge_sum(const int* __restrict__ src, const int* __restrict__ dst,
                           const float* __restrict__ z, float* __restrict__ agg, int E) {
  int wid = (int)(((long)blockIdx.x * blockDim.x + threadIdx.x) >> 5);
  int lane = threadIdx.x & 31;
  if (wid >= E) return;
  int s = src[wid], d = dst[wid];
  float4 zv = ((const float4*)(z + (long)s * HCH))[lane];
  float* ad = agg + (long)d * HCH + lane * 4;
  atomicAdd(ad + 0, zv.x);
  atomicAdd(ad + 1, zv.y);
  atomicAdd(ad + 2, zv.z);
  atomicAdd(ad + 3, zv.w);
}

// z += h*dinv^2 + b0 ; x0 = z   (epilogue of initial GCNConv)
__global__ void k_self_bias(const float* __restrict__ h, const float* __restrict__ dinv,
                            const float* __restrict__ b0, float* __restrict__ z,
                            float* __restrict__ x0, int N) {
  long i = (long)blockIdx.x * blockDim.x + threadIdx.x;
  long n = (long)N * HCH;
  if (i >= n) return;
  int r = (int)(i >> 7);
  int c = (int)(i & (HCH - 1));
  float di = dinv[r];
  float v = z[i] + h[i] * di * di + b0[c];
  z[i] = v;
  x0[i] = v;
}

// ---------------- WMMA core ----------------
// A 16x4 f32: lanes 0-15 -> K 0..1, lanes 16-31 -> K 2..3 (2 VGPRs)
// B 4x16 f32: lanes select column; half-wave selects K pair (2 VGPRs)
// C/D 16x16 f32: VGPR v holds row v (lanes 0-15) / row v+8 (lanes 16-31), col = lane&15
__device__ __forceinline__ v8f wmma_rowtile(const float* __restrict__ sX,
                                            const float* __restrict__ sW,
                                            int lrow0, int nt, int lane) {
  v8f acc = {0.f, 0.f, 0.f, 0.f, 0.f, 0.f, 0.f, 0.f};
  const int m  = lrow0 + (lane & 15);
  const int kh = (lane >> 4) << 1;       // 0 or 2
  const int n  = nt * 16 + (lane & 15);
#pragma unroll
  for (int k = 0; k < HCH; k += 4) {
    v2f a, b;
    a.x = sX[m * HCH + k + kh];
    a.y = sX[m * HCH + k + kh + 1];
    b.x = sW[(k + kh) * HCH + n];
    b.y = sW[(k + kh + 1) * HCH + n];
    acc = __builtin_amdgcn_wmma_f32_16x16x4_f32(false, a, false, b,
                                                (short)0, acc, false, false);
  }
  return acc;
}

// ---------------- plain GEMM: OUT = X @ W  (h = x @ w0) ----------------
__global__ void k_wmma_gemm(const float* __restrict__ X, const float* __restrict__ W,
                            float* __restrict__ OUT, int N) {
  extern __shared__ float lds[];
  float* sW = lds;                 // 128x128
  float* sX = lds + HCH * HCH;     // 64x128
  const int tid = threadIdx.x, lane = tid & 31, wave = tid >> 5;
  const long rowBase = (long)blockIdx.x * RPB;

  const float4* Wv = (const float4*)W;
  float4* sWv = (float4*)sW;
  for (int i = tid; i < HCH * HCH / 4; i += TPB) sWv[i] = Wv[i];

  for (int i = tid; i < RPB * HCH / 4; i += TPB) {
    int r = i / (HCH / 4);
    long gr = rowBase + r;
    float4 v = {0.f, 0.f, 0.f, 0.f};
    if (gr < N) v = ((const float4*)X)[gr * (HCH / 4) + (i % (HCH / 4))];
    ((float4*)sX)[i] = v;
  }
  __syncthreads();

  const int lrow0 = wave * 16;
#pragma unroll
  for (int nt = 0; nt < 8; ++nt) {
    v8f acc = wmma_rowtile(sX, sW, lrow0, nt, lane);
    const int n = nt * 16 + (lane & 15);
    const int mh = (lane >> 4) * 8;
#pragma unroll
    for (int v = 0; v < 8; ++v) {
      long gr = rowBase + lrow0 + mh + v;
      if (gr < N) OUT[gr * HCH + n] = acc[v];
    }
  }
}

// ------------- fused GCN2Conv layer: x1 = (1-a)agg + a*x0; p = x1@W;
//               zpre = (1-b)x1 + b*p; JK running max; BN(eval)+ReLU -------------
__global__ void k_wmma_layer(const float* __restrict__ agg, const float* __restrict__ x0,
                             const float* __restrict__ W, const float* __restrict__ gamma,
                             const float* __restrict__ bnb,
                             float* __restrict__ z, float* __restrict__ jk,
                             float* __restrict__ x0_out,
                             float beta, float alpha, float bn_inv,
                             int isStart, int isEnd, int N) {
  extern __shared__ float lds[];
  float* sW = lds;
  float* sX = lds + HCH * HCH;
  const int tid = threadIdx.x, lane = tid & 31, wave = tid >> 5;
  const long rowBase = (long)blockIdx.x * RPB;

  const float4* Wv = (const float4*)W;
  float4* sWv = (float4*)sW;
  for (int i = tid; i < HCH * HCH / 4; i += TPB) sWv[i] = Wv[i];

  const float oma = 1.0f - alpha;
  for (int i = tid; i < RPB * HCH / 4; i += TPB) {
    int r = i / (HCH / 4);
    long gr = rowBase + r;
    float4 v = {0.f, 0.f, 0.f, 0.f};
    if (gr < N) {
      long o = gr * (HCH / 4) + (i % (HCH / 4));
      float4 a4 = ((const float4*)agg)[o];
      float4 x4 = ((const float4*)x0)[o];
      v.x = oma * a4.x + alpha * x4.x;
      v.y = oma * a4.y + alpha * x4.y;
      v.z = oma * a4.z + alpha * x4.z;
      v.w = oma * a4.w + alpha * x4.w;
    }
    ((float4*)sX)[i] = v;
  }
  __syncthreads();

  const int lrow0 = wave * 16;
  const float omb = 1.0f - beta;
#pragma unroll
  for (int nt = 0; nt < 8; ++nt) {
    v8f acc = wmma_rowtile(sX, sW, lrow0, nt, lane);
    const int n = nt * 16 + (lane & 15);
    const int mh = (lane >> 4) * 8;
    const float gsc = gamma[n] * bn_inv;
    const float gbb = bnb[n];
#pragma unroll
    for (int v = 0; v < 8; ++v) {
      int lr = lrow0 + mh + v;
      long gr = rowBase + lr;
      if (gr < N) {
        float x1v  = sX[lr * HCH + n];
        float zpre = omb * x1v + beta * acc[v];
        long  o    = gr * HCH + n;
        float jkv  = isStart ? zpre : fmaxf(jk[o], zpre);
        jk[o] = jkv;
        if (isEnd) {
          z[o] = jkv;        // JK 'max' replaces activation at group end
          x0_out[o] = jkv;   // and becomes the new residual anchor
        } else {
          z[o] = fmaxf(zpre * gsc + gbb, 0.f);
        }
      }
    }
  }
}

// ---------------- host launcher ----------------
extern "C" void kernel_launch(void* const* d_in, const int* in_sizes, int n_in,
                              void* d_out, int out_size, void* d_ws, size_t ws_size,
                              hipStream_t stream) {
  const float* x      = (const float*)d_in[0];   // [N,H]
  const float* w0     = (const float*)d_in[1];   // [H,H]
  const float* b0     = (const float*)d_in[2];   // [H]
  const float* conv_w = (const float*)d_in[3];   // [L,H,H]
  const float* gamma  = (const float*)d_in[4];   // [L,H]
  const float* bnb    = (const float*)d_in[5];   // [L,H]
  const int*   eidx   = (const int*)d_in[6];     // [2,E]

  const int N = in_sizes[0] / HCH;
  const int E = in_sizes[6] / 2;
  const int L = in_sizes[3] / (HCH * HCH);
  const int* src = eidx;
  const int* dst = eidx + E;

  const long  NH = (long)N * HCH;
  float* ws  = (float*)d_ws;
  float* deg = ws;                                  // N floats (deg -> dinv)
  float* buf = ws + (((long)N + 255) & ~255L);      // N*H  (h, then agg per layer)
  float* x0  = buf + NH;                            // N*H
  float* jk  = x0 + NH;                             // N*H
  float* z   = (float*)d_out;                       // N*H, final JK-max lands here

  const int thr = 256;
  const int nbGemm  = (N + RPB - 1) / RPB;
  const int nbElem  = (int)((NH + thr - 1) / thr);
  const int nbEdgeT = (E + thr - 1) / thr;
  const int nbEdgeW = (E + (thr / 32) - 1) / (thr / 32);  // one wave per edge

  // degrees: deg = 1 + sum over incoming edges; dinv = rsqrt(deg)
  k_fill<<<(N + thr - 1) / thr, thr, 0, stream>>>(deg, 1.0f, (long)N);
  k_deg_count<<<nbEdgeT, thr, 0, stream>>>(dst, deg, E);
  k_rsqrt<<<(N + thr - 1) / thr, thr, 0, stream>>>(deg, N);

  // h = x @ w0
  k_wmma_gemm<<<nbGemm, TPB, LDS_BYTES, stream>>>(x, w0, buf, N);

  // z = segsum(h[src]*dinv_s*dinv_d, dst) + h*dinv^2 + b0 ; x0 = z
  k_fill<<<nbElem, thr, 0, stream>>>(z, 0.0f, NH);
  k_edge_gcn<<<nbEdgeW, thr, 0, stream>>>(src, dst, deg, buf, z, E);
  k_self_bias<<<nbElem, thr, 0, stream>>>(buf, deg, b0, z, x0, N);

  const float alpha  = 0.1f;
  const float bn_inv = 1.0f / sqrtf(1.0f + 1e-5f);

  for (int i = 0; i < L; ++i) {
    const float beta = logf(0.5f / (float)(i + 1) + 1.0f);
    // agg = segsum(z[src], dst)
    k_fill<<<nbElem, thr, 0, stream>>>(buf, 0.0f, NH);
    k_edge_sum<<<nbEdgeW, thr, 0, stream>>>(src, dst, z, buf, E);
    // fused combine + GEMM + JK + BN/ReLU
    k_wmma_layer<<<nbGemm, TPB, LDS_BYTES, stream>>>(
        buf, x0, conv_w + (long)i * HCH * HCH, gamma + (long)i * HCH,
        bnb + (long)i * HCH, z, jk, x0,
        beta, alpha, bn_inv,
        (i % 4 == 0) ? 1 : 0, (i % 4 == 3) ? 1 : 0, N);
  }
}